// RBFLayer_25417616458394
// MI455X (gfx1250) — compile-verified
//
#include <hip/hip_runtime.h>

#define IN_DIM      128
#define NUM_KERNELS 512
#define OUT_DIM     256
#define EPS_NORM    1e-9f

#define ROWS       16            // rows of x per workgroup
#define SX_STRIDE  (IN_DIM + 4)  // 132: 132%64==4 -> conflict-free column reads
#define SR_STRIDE  (NUM_KERNELS + 4) // 516: halves of C-reg writes hit disjoint banks

typedef __attribute__((ext_vector_type(2))) float v2f;
typedef __attribute__((ext_vector_type(8))) float v8f;

// Precompute cn2[k] = ||C_k||^2 and eps2[k] = exp(2*log_shape[k]) into workspace.
__global__ void rbf_precompute(const float* __restrict__ centers,
                               const float* __restrict__ log_shapes,
                               float* __restrict__ cn2,
                               float* __restrict__ eps2) {
    int k = blockIdx.x * blockDim.x + threadIdx.x;
    if (k >= NUM_KERNELS) return;
    const float* c = centers + k * IN_DIM;
    float s = 0.f;
#pragma unroll 8
    for (int d = 0; d < IN_DIM; ++d) { float v = c[d]; s += v * v; }
    cn2[k]  = s;
    eps2[k] = __expf(2.0f * log_shapes[k]);
}

// Fused: distance-GEMM (WMMA f32) -> gaussian -> row-sum -> output-GEMM (WMMA f32),
// with normalization folded into the final row scaling.
__global__ __launch_bounds__(256) void rbf_fused(
    const float* __restrict__ x,        // N x 128 row-major
    const float* __restrict__ centers,  // 512 x 128 row-major
    const float* __restrict__ weights,  // 256 x 512 row-major
    const float* __restrict__ cn2,      // 512
    const float* __restrict__ eps2,     // 512
    float* __restrict__ out)            // N x 256
{
    __shared__ float sX[ROWS][SX_STRIDE];   // ~8.4 KB
    __shared__ float sR[ROWS][SR_STRIDE];   // ~33 KB rbf tile
    __shared__ float sXn2[ROWS];
    __shared__ float sSum[ROWS];
    __shared__ float sInv[ROWS];

    const int tid     = threadIdx.x;
    const int wave    = tid >> 5;      // 0..7 (wave32)
    const int lane    = tid & 31;
    const int rowbase = blockIdx.x * ROWS;

    // Load X tile (16x128) into LDS.
    {
        const float* xg = x + (size_t)rowbase * IN_DIM;
        for (int i = tid; i < ROWS * IN_DIM; i += 256) {
            int r = i >> 7, d = i & 127;
            sX[r][d] = xg[i];
        }
    }
    if (tid < ROWS) sSum[tid] = 0.f;
    __syncthreads();
    if (tid < ROWS) {
        float s = 0.f;
        for (int d = 0; d < IN_DIM; ++d) { float v = sX[tid][d]; s += v * v; }
        sXn2[tid] = s;
    }
    __syncthreads();

    // WMMA f32 16x16x4 lane mapping (A and B mirror each other):
    //   m_or_n = lane & 15 ; koff = (lane>>4)*2 ; regs hold {koff, koff+1}
    const int colh = lane & 15;
    const int koff = (lane >> 4) << 1;

    // ---- Phase B: S = X @ C^T over this wave's 64 K-columns, 4 tiles of 16x16
    for (int t = 0; t < 4; ++t) {
        const int k0 = wave * 64 + t * 16;
        v8f acc = {};
#pragma unroll 4
        for (int d0 = 0; d0 < IN_DIM; d0 += 4) {
            v2f a, b;
            a.x = sX[colh][d0 + koff];
            a.y = sX[colh][d0 + koff + 1];
            const float* cp = centers + (size_t)(k0 + colh) * IN_DIM + d0 + koff;
            b.x = cp[0];
            b.y = cp[1];
            acc = __builtin_amdgcn_wmma_f32_16x16x4_f32(
                false, a, false, b, (short)0, acc, false, false);
        }
        // C/D layout: reg j -> row (lane<16 ? j : j+8), col = colh
#pragma unroll
        for (int j = 0; j < 8; ++j) {
            int row   = (lane < 16) ? j : (j + 8);
            int k     = k0 + colh;
            float dot = acc[j];
            float r2  = sXn2[row] - 2.0f * dot + cn2[k];
            r2        = fmaxf(r2, 0.0f);
            float val = __expf(-eps2[k] * r2);
            sR[row][k] = val;
            atomicAdd(&sSum[row], val);   // ds_add_f32
        }
    }
    __syncthreads();
    if (tid < ROWS) sInv[tid] = 1.0f / (sSum[tid] + EPS_NORM);
    __syncthreads();

    // ---- Phase D: out = rbf @ W^T ; two 16-wide output tiles per wave.
    for (int t = 0; t < 2; ++t) {
        const int n0 = (wave * 2 + t) * 16;  // 0..240
        v8f acc = {};
#pragma unroll 4
        for (int k0 = 0; k0 < NUM_KERNELS; k0 += 4) {
            v2f a, b;
            a.x = sR[colh][k0 + koff];
            a.y = sR[colh][k0 + koff + 1];
            const float* wp = weights + (size_t)(n0 + colh) * NUM_KERNELS + k0 + koff;
            b.x = wp[0];
            b.y = wp[1];
            acc = __builtin_amdgcn_wmma_f32_16x16x4_f32(
                false, a, false, b, (short)0, acc, false, false);
        }
#pragma unroll
        for (int j = 0; j < 8; ++j) {
            int row = (lane < 16) ? j : (j + 8);
            out[(size_t)(rowbase + row) * OUT_DIM + n0 + colh] = acc[j] * sInv[row];
        }
    }
}

extern "C" void kernel_launch(void* const* d_in, const int* in_sizes, int n_in,
                              void* d_out, int out_size, void* d_ws, size_t ws_size,
                              hipStream_t stream) {
    const float* x          = (const float*)d_in[0];   // 16*512*128
    const float* centers    = (const float*)d_in[1];   // 512*128
    const float* log_shapes = (const float*)d_in[2];   // 512
    const float* weights    = (const float*)d_in[3];   // 256*512
    float* out = (float*)d_out;

    float* cn2  = (float*)d_ws;          // 512 floats
    float* eps2 = cn2 + NUM_KERNELS;     // 512 floats

    const int N = in_sizes[0] / IN_DIM;  // 8192

    rbf_precompute<<<(NUM_KERNELS + 255) / 256, 256, 0, stream>>>(
        centers, log_shapes, cn2, eps2);
    rbf_fused<<<N / ROWS, 256, 0, stream>>>(
        x, centers, weights, cn2, eps2, out);
}